// TKANCell_23519240913408
// MI455X (gfx1250) — compile-verified
//
#include <hip/hip_runtime.h>
#include <hip/hip_bf16.h>
#include <math.h>

#define Bc   128
#define Tc   256
#define Dc   256
#define Uc   256
#define Sc   3
#define ZC   768            // 3*U
#define ROWS 16             // batch rows per WG (WMMA M)
#define NWG  (Bc / ROWS)    // 8 workgroups
#define BLK  512            // 16 waves/WG
#define WAVES (BLK / 32)
#define CTILES 48           // 768 / 16 column tiles
#define TPW  (CTILES / WAVES)  // 3 col tiles per wave
#define KIT  128            // K iterations: 512 / 4
#define XP   260            // padded row stride for x/h in LDS (bank-conflict-free)
#define ZP   772            // padded row stride for hz in LDS

typedef __attribute__((ext_vector_type(2))) float v2f;
typedef __attribute__((ext_vector_type(8))) float v8f;

#define BSW_ELEMS (CTILES * KIT * 32)   // v2f elements: 48*128*32 (1.5 MB)

__device__ __forceinline__ float sigmoidf_(float v) {
    return 1.0f / (1.0f + __expf(-v));
}

// cubic B-spline basis on uniform grid (-2.2 .. 2.2, h=0.4), dotted with 8 coeffs
__device__ __forceinline__ float spline_dot(float xv, const float* __restrict__ sw) {
    float b[11];
#pragma unroll
    for (int j = 0; j < 11; ++j) {
        float gj = -2.2f + 0.4f * (float)j;
        b[j] = (xv >= gj && xv < gj + 0.4f) ? 1.0f : 0.0f;
    }
#pragma unroll
    for (int k = 1; k <= 3; ++k) {
        float inv = 1.0f / (0.4f * (float)k);
#pragma unroll
        for (int j = 0; j < 11 - k; ++j) {
            float gj = -2.2f + 0.4f * (float)j;
            b[j] = (xv - gj) * inv * b[j] +
                   ((gj + 0.4f * (float)(k + 1)) - xv) * inv * b[j + 1];
        }
    }
    float acc = 0.0f;
#pragma unroll
    for (int cc = 0; cc < 8; ++cc) acc += b[cc] * sw[cc];
    return acc;
}

// one-time: pre-swizzle [wk; wr] into per-lane WMMA B-operand layout
// bsw[tile][it][lane] = { W[k][n], W[k+1][n] }, k = 4*it + 2*(lane>>4),
// n = 16*tile + (lane&15)
__global__ void tkan_bswz(const float* __restrict__ wk,
                          const float* __restrict__ wr,
                          float* __restrict__ bsw) {
    int e = blockIdx.x * 256 + threadIdx.x;
    if (e >= BSW_ELEMS) return;
    const int lane = e & 31;
    const int it   = (e >> 5) & (KIT - 1);
    const int tile = e >> 12;
    const int k    = 4 * it + 2 * (lane >> 4);
    const int n    = tile * 16 + (lane & 15);
    float v0, v1;
    if (k < Dc) {
        v0 = wk[(size_t)k * ZC + n];
        v1 = wk[(size_t)(k + 1) * ZC + n];
    } else {
        v0 = wr[(size_t)(k - Dc) * ZC + n];
        v1 = wr[(size_t)(k + 1 - Dc) * ZC + n];
    }
    bsw[(size_t)e * 2 + 0] = v0;
    bsw[(size_t)e * 2 + 1] = v1;
}

__global__ __launch_bounds__(BLK) void tkan_scan(
    const float* __restrict__ x,       // (B,T,D)
    const float* __restrict__ bias,    // (3U)
    const float* __restrict__ subk,    // (S,2)
    const float* __restrict__ subrk,   // (S,2D)
    const float* __restrict__ aggw,    // (S,U)
    const float* __restrict__ aggb,    // (U)
    const float* __restrict__ kbase,   // (S,D,1)
    const float* __restrict__ kspl,    // (S,D,8,1)
    const float* __restrict__ bsw,     // pre-swizzled B operands
    float* __restrict__ out)           // (B,T,U)
{
    __shared__ float xs[ROWS * XP];    // staged x_t tile
    __shared__ float hs[ROWS * XP];    // recurrent h tile
    __shared__ float hzs[ROWS * ZP];   // z = [x_t,h]@[wk;wr] for this tile
    __shared__ float subs[Sc * ROWS];  // KAN recurrent state
    __shared__ float subos[Sc * ROWS]; // KAN output

    const int rbase = blockIdx.x * ROWS;       // global batch row base
    const int lane  = threadIdx.x & 31;
    const int wave  = threadIdx.x >> 5;        // 0..15
    const int l15   = lane & 15;
    const int hf    = lane >> 4;               // 0: K=k0,k0+1  1: K=k0+2,k0+3
    const int orow  = 8 * hf;                  // D-matrix base row for this lane

    const v2f* bswv = (const v2f*)bsw;

    // zero recurrent state
    for (int i = threadIdx.x; i < ROWS * XP; i += BLK) hs[i] = 0.0f;
    if (threadIdx.x < Sc * ROWS) subs[threadIdx.x] = 0.0f;
    float c_reg[8];
#pragma unroll
    for (int j = 0; j < 8; ++j) c_reg[j] = 0.0f;
    __syncthreads();

    for (int t = 0; t < Tc; ++t) {
        // ---- stage x_t tile into LDS (coalesced) ----
        for (int i = threadIdx.x; i < ROWS * Dc; i += BLK) {
            const int r = i >> 8;
            const int d = i & 255;
            xs[r * XP + d] = x[((size_t)(rbase + r) * Tc + t) * Dc + d];
        }
        __syncthreads();

        // ---- Phase A: GEMM  z = [x_t, h] @ [wk; wr]  (WMMA f32) ----
        v8f acc0 = {}, acc1 = {}, acc2 = {};
        const int t0 = wave * TPW;             // this wave's first col tile
        const v2f* b0p = bswv + (size_t)(t0 + 0) * KIT * 32;
        const v2f* b1p = bswv + (size_t)(t0 + 1) * KIT * 32;
        const v2f* b2p = bswv + (size_t)(t0 + 2) * KIT * 32;
        const float* xr = xs + l15 * XP + 2 * hf;
        const float* hr = hs + l15 * XP + 2 * hf;
#pragma unroll 4
        for (int it = 0; it < KIT / 2; ++it) {          // K = x part
            v2f a = *(const v2f*)(xr + 4 * it);
            v2f b0 = b0p[it * 32 + lane];
            v2f b1 = b1p[it * 32 + lane];
            v2f b2 = b2p[it * 32 + lane];
            acc0 = __builtin_amdgcn_wmma_f32_16x16x4_f32(false, a, false, b0,
                                                         (short)0, acc0, false, false);
            acc1 = __builtin_amdgcn_wmma_f32_16x16x4_f32(false, a, false, b1,
                                                         (short)0, acc1, false, false);
            acc2 = __builtin_amdgcn_wmma_f32_16x16x4_f32(false, a, false, b2,
                                                         (short)0, acc2, false, false);
        }
#pragma unroll 4
        for (int it = KIT / 2; it < KIT; ++it) {        // K = h part
            v2f a = *(const v2f*)(hr + 4 * (it - KIT / 2));
            v2f b0 = b0p[it * 32 + lane];
            v2f b1 = b1p[it * 32 + lane];
            v2f b2 = b2p[it * 32 + lane];
            acc0 = __builtin_amdgcn_wmma_f32_16x16x4_f32(false, a, false, b0,
                                                         (short)0, acc0, false, false);
            acc1 = __builtin_amdgcn_wmma_f32_16x16x4_f32(false, a, false, b1,
                                                         (short)0, acc1, false, false);
            acc2 = __builtin_amdgcn_wmma_f32_16x16x4_f32(false, a, false, b2,
                                                         (short)0, acc2, false, false);
        }
        {
            const int n0 = t0 * 16 + l15;
#pragma unroll
            for (int j = 0; j < 8; ++j) {
                hzs[(orow + j) * ZP + n0]      = acc0[j];
                hzs[(orow + j) * ZP + n0 + 16] = acc1[j];
                hzs[(orow + j) * ZP + n0 + 32] = acc2[j];
            }
        }

        // ---- Phase A: KAN sub-network (3 rows per wave, 48 rows total) ----
#pragma unroll
        for (int rr = 0; rr < TPW; ++rr) {
            const int row  = wave * TPW + rr;   // 0..47  (s*16 + local row)
            const int s    = row >> 4;
            const int rloc = row & 15;
            const float subv = subs[row];
            const float* skx = subrk + (size_t)s * (2 * Dc);
            const float* skh = skx + Dc;
            const float* bw  = kbase + (size_t)s * Dc;
            const float* swl = kspl + (size_t)s * Dc * 8;
            float acc = 0.0f;
            for (int d = lane; d < Dc; d += 32) {
                float xv = xs[rloc * XP + d];
                float a  = xv * skx[d] + subv * skh[d];
                acc += a * sigmoidf_(a) * bw[d];          // silu(a)*base_w
                acc += spline_dot(a, swl + (size_t)d * 8);
            }
#pragma unroll
            for (int off = 16; off > 0; off >>= 1)
                acc += __shfl_xor(acc, off, 32);
            if (lane == 0) subos[row] = acc;
        }
        __syncthreads();

        // ---- Phase B: gates, state update, output (8 elements/thread) ----
#pragma unroll
        for (int j = 0; j < 8; ++j) {
            const int idx = threadIdx.x + j * BLK;   // 0..4095
            const int r = idx >> 8;
            const int u = idx & 255;
            const float* hzr = hzs + r * ZP;
            float zi = hzr[u]           + bias[u];
            float zf = hzr[Uc + u]      + bias[Uc + u];
            float zc = hzr[2 * Uc + u]  + bias[2 * Uc + u];
            float iv = sigmoidf_(zi);
            float fv = sigmoidf_(zf);
            float cn = fv * c_reg[j] + iv * tanhf(zc);
            float og = aggb[u]
                     + subos[r]             * aggw[u]
                     + subos[ROWS + r]      * aggw[Uc + u]
                     + subos[2 * ROWS + r]  * aggw[2 * Uc + u];
            float ov = sigmoidf_(og);
            float hn = ov * tanhf(cn);
            c_reg[j] = cn;
            hs[r * XP + u] = hn;
            out[((size_t)(rbase + r) * Tc + t) * Uc + u] = hn;
        }
        if (threadIdx.x < Sc * ROWS) {
            const int s = threadIdx.x >> 4;
            subs[threadIdx.x] = subk[s * 2 + 0] * subos[threadIdx.x]
                              + subk[s * 2 + 1] * subs[threadIdx.x];
        }
        __syncthreads();
    }
}

extern "C" void kernel_launch(void* const* d_in, const int* in_sizes, int n_in,
                              void* d_out, int out_size, void* d_ws, size_t ws_size,
                              hipStream_t stream) {
    const float* x     = (const float*)d_in[0];
    const float* wk    = (const float*)d_in[1];
    const float* wr    = (const float*)d_in[2];
    const float* bias  = (const float*)d_in[3];
    const float* subk  = (const float*)d_in[4];
    const float* subrk = (const float*)d_in[5];
    const float* aggw  = (const float*)d_in[6];
    const float* aggb  = (const float*)d_in[7];
    const float* kbase = (const float*)d_in[8];
    const float* kspl  = (const float*)d_in[9];
    float* out = (float*)d_out;
    float* bsw = (float*)d_ws;   // 1.5 MB pre-swizzled weights

    tkan_bswz<<<(BSW_ELEMS + 255) / 256, 256, 0, stream>>>(wk, wr, bsw);

    tkan_scan<<<NWG, BLK, 0, stream>>>(x, bias, subk, subrk,
                                       aggw, aggb, kbase, kspl, bsw, out);
}